// PropLayer_69887707840823
// MI455X (gfx1250) — compile-verified
//
#include <hip/hip_runtime.h>
#include <stdint.h>

#define D 256
#define WAVES_PER_BLOCK 8
#define PIPE_DEPTH 4

typedef float vfloat4 __attribute__((ext_vector_type(4)));

// GCC-style int4 vector: matches the builtin's 'V4i' parameter type exactly.
typedef int v4i_gcc __attribute__((vector_size(16)));
typedef __attribute__((address_space(1))) v4i_gcc* as1_v4i;
typedef __attribute__((address_space(3))) v4i_gcc* as3_v4i;

#if defined(__AMDGCN__) && __has_builtin(__builtin_amdgcn_global_load_async_to_lds_b128) && __has_builtin(__builtin_amdgcn_s_wait_asynccnt)
#define HAVE_ASYNC_LDS 1
#else
#define HAVE_ASYNC_LDS 0
#endif

// ---------------------------------------------------------------------------
// Output init: stream-copy paper_h into out[0:paper_elems) with non-temporal
// hints (don't pollute L2 -- the scatter accumulators want to live there),
// zero the author/venue accumulator regions with regular (L2-resident) stores.
// ---------------------------------------------------------------------------
__global__ __launch_bounds__(256) void init_out_kernel(const float* __restrict__ paper_h,
                                                       float* __restrict__ out,
                                                       size_t paper_elems,
                                                       size_t total_elems) {
  size_t i = ((size_t)blockIdx.x * blockDim.x + threadIdx.x) * 4;
  const size_t stride = (size_t)gridDim.x * blockDim.x * 4;
  for (; i < total_elems; i += stride) {
    if (i < paper_elems) {
      vfloat4 v = __builtin_nontemporal_load((const vfloat4*)(paper_h + i));
      __builtin_nontemporal_store(v, (vfloat4*)(out + i));
    } else {
      vfloat4 z = {0.f, 0.f, 0.f, 0.f};
      *(vfloat4*)(out + i) = z;
    }
  }
}

__device__ __forceinline__ void atomic_add4(float* o, vfloat4 v) {
  (void)__hip_atomic_fetch_add(o + 0, v.x, __ATOMIC_RELAXED, __HIP_MEMORY_SCOPE_AGENT);
  (void)__hip_atomic_fetch_add(o + 1, v.y, __ATOMIC_RELAXED, __HIP_MEMORY_SCOPE_AGENT);
  (void)__hip_atomic_fetch_add(o + 2, v.z, __ATOMIC_RELAXED, __HIP_MEMORY_SCOPE_AGENT);
  (void)__hip_atomic_fetch_add(o + 3, v.w, __ATOMIC_RELAXED, __HIP_MEMORY_SCOPE_AGENT);
}

#if HAVE_ASYNC_LDS
// Each lane async-copies 16B at [lane*4] and [128+lane*4] of a 256-float row:
// 2 wave-level instructions move the full 1KB row, tracked on ASYNCcnt.
__device__ __forceinline__ void async_copy_row_b128(const float* grow, float* lrow, int lane) {
  __builtin_amdgcn_global_load_async_to_lds_b128(
      (as1_v4i)(grow + lane * 4), (as3_v4i)(lrow + lane * 4), 0, 0);
  __builtin_amdgcn_global_load_async_to_lds_b128(
      (as1_v4i)(grow + 128 + lane * 4), (as3_v4i)(lrow + 128 + lane * 4), 0, 0);
}
#endif

// ---------------------------------------------------------------------------
// Edge scatter: one wave per edge iteration, 8 floats per lane.
//   out[dst] += src_h[src] * w1[e] (* w2[e] if w2 != null)
// Async path: PIPE_DEPTH edge-rows in flight per wave through LDS, consumed
// in order with precise s_wait_asynccnt thresholds (async loads complete
// in order, 2 instructions per edge).
// ---------------------------------------------------------------------------
__global__ __launch_bounds__(256) void scatter_edges_kernel(
    const float* __restrict__ src_h, const int* __restrict__ src_idx,
    const int* __restrict__ dst_idx, const float* __restrict__ w1,
    const float* __restrict__ w2, float* __restrict__ out_base, int n_edges) {
  const int lane = threadIdx.x & 31;
  const int wv = threadIdx.x >> 5;
  const long gw = (long)blockIdx.x * WAVES_PER_BLOCK + wv;
  const long nw = (long)gridDim.x * WAVES_PER_BLOCK;
  const long E = (long)n_edges;

#if HAVE_ASYNC_LDS
  __shared__ __align__(16) float buf[WAVES_PER_BLOCK * PIPE_DEPTH * D];  // 32 KB
  float* mybuf = &buf[wv * PIPE_DEPTH * D];

  long issued = 0;  // edges issued (this wave)
  for (int d = 0; d < PIPE_DEPTH; ++d) {
    const long e = gw + (long)d * nw;
    if (e < E) {
      const int s = src_idx[e];
      async_copy_row_b128(src_h + (size_t)s * D, mybuf + d * D, lane);
      ++issued;
    }
  }

  long i = 0;
  for (long e = gw; e < E; e += nw, ++i) {
    const int slot = (int)(i & (PIPE_DEPTH - 1));
    const long pending = issued - i;  // edges in flight, incl. this one (>=1)
    // Allow 2*(pending-1) outstanding async instrs => edge i's 2 loads landed
    // (async loads return "done" in order; 2 instructions per edge).
    if (pending <= 1)      __builtin_amdgcn_s_wait_asynccnt(0);
    else if (pending == 2) __builtin_amdgcn_s_wait_asynccnt(2);
    else if (pending == 3) __builtin_amdgcn_s_wait_asynccnt(4);
    else                   __builtin_amdgcn_s_wait_asynccnt(6);

    float w = w1[e];
    if (w2) w *= w2[e];
    const int dv = dst_idx[e];
    const float* lrow = mybuf + slot * D;
    vfloat4 a = *(const vfloat4*)(lrow + lane * 4);
    vfloat4 b = *(const vfloat4*)(lrow + 128 + lane * 4);

    float* o = out_base + (size_t)dv * D + lane * 4;
    atomic_add4(o, a * w);
    atomic_add4(o + 128, b * w);

    // Refill this slot with the next edge for this wave.
    const long en = gw + issued * nw;
    if (en < E) {
      const int s = src_idx[en];
      async_copy_row_b128(src_h + (size_t)s * D, mybuf + slot * D, lane);
      ++issued;
    }
  }
#else
  for (long e = gw; e < E; e += nw) {
    const long en = e + nw;
    if (en < E) __builtin_prefetch(&src_idx[en], 0, 0);  // global_prefetch_b8
    const int s = src_idx[e];
    const float* row = src_h + (size_t)s * D + lane * 4;
    vfloat4 a = *(const vfloat4*)row;
    vfloat4 b = *(const vfloat4*)(row + 128);
    float w = w1[e];
    if (w2) w *= w2[e];
    const int dv = dst_idx[e];
    float* o = out_base + (size_t)dv * D + lane * 4;
    atomic_add4(o, a * w);
    atomic_add4(o + 128, b * w);
  }
#endif
}

// ---------------------------------------------------------------------------
// Host-side launch. Inputs (setup_inputs order):
// 0 paper_h  1 author_h  2 wb_src  3 wb_dst  4 wb_w  5 pi_src  6 pi_dst
// 7 pi_alpha 8 pi_w  9 hp_src  10 hp_dst  11 hp_feat  12 hp_alpha  13 n_venues
// Output: [paper_h | author_out | venue_out] flattened float32.
// ---------------------------------------------------------------------------
extern "C" void kernel_launch(void* const* d_in, const int* in_sizes, int n_in,
                              void* d_out, int out_size, void* d_ws, size_t ws_size,
                              hipStream_t stream) {
  const float* paper_h  = (const float*)d_in[0];
  const float* author_h = (const float*)d_in[1];
  const int*   wb_src   = (const int*)d_in[2];
  const int*   wb_dst   = (const int*)d_in[3];
  const float* wb_w     = (const float*)d_in[4];
  const int*   pi_src   = (const int*)d_in[5];
  const int*   pi_dst   = (const int*)d_in[6];
  const float* pi_alpha = (const float*)d_in[7];
  const float* pi_w     = (const float*)d_in[8];
  const int*   hp_src   = (const int*)d_in[9];
  const int*   hp_dst   = (const int*)d_in[10];
  const float* hp_feat  = (const float*)d_in[11];
  const float* hp_alpha = (const float*)d_in[12];

  const size_t paper_elems  = (size_t)in_sizes[0];
  const size_t author_elems = (size_t)in_sizes[1];
  const size_t total_elems  = (size_t)out_size;

  float* out        = (float*)d_out;
  float* author_out = out + paper_elems;
  float* venue_out  = out + paper_elems + author_elems;

  {
    const size_t vec = total_elems / 4;
    long blocks = (long)((vec + 255) / 256);
    if (blocks > 262144) blocks = 262144;
    if (blocks < 1) blocks = 1;
    init_out_kernel<<<dim3((unsigned)blocks), dim3(256), 0, stream>>>(
        paper_h, out, paper_elems, total_elems);
  }

  auto launch_scatter = [&](const float* sh, const int* si, const int* di,
                            const float* w1, const float* w2, float* ob, int E) {
    if (E <= 0) return;
    long blocks = ((long)E + 255) / 256;  // -> ~32 edges per wave at the cap
    if (blocks > 4096) blocks = 4096;
    scatter_edges_kernel<<<dim3((unsigned)blocks), dim3(256), 0, stream>>>(
        sh, si, di, w1, w2, ob, E);
  };

  // writtenby: paper -> author, w = wb_w
  launch_scatter(paper_h,  wb_src, wb_dst, wb_w,     nullptr,  author_out, in_sizes[2]);
  // publishedin: paper -> venue, w = pi_alpha * pi_w
  launch_scatter(paper_h,  pi_src, pi_dst, pi_alpha, pi_w,     venue_out,  in_sizes[5]);
  // haspaperin: author -> venue, w = hp_feat * hp_alpha
  launch_scatter(author_h, hp_src, hp_dst, hp_feat,  hp_alpha, venue_out,  in_sizes[9]);
}